// Memory_Module_21723944583255
// MI455X (gfx1250) — compile-verified
//
#include <hip/hip_runtime.h>
#include <stdint.h>

// ---------------- sizes (BATCH = 16, fixed by the reference) ----------------
static constexpr size_t N_B = 16ull * 256 * 32 * 32;    //  4,194,304 f32 (16 MB)
static constexpr size_t N_3 = 16ull * 128 * 64 * 64;    //  8,388,608 f32 (32 MB)
static constexpr size_t N_2 = 16ull * 64 * 128 * 128;   // 16,777,216 f32 (64 MB)
static constexpr size_t N_1 = 16ull * 64 * 256 * 256;   // 67,108,864 f32 (256 MB)

#define COPY_BLOCK  256                       // threads per block (8 waves)
#define COPY_BLOCKS 1024                      // fixed grid (compile-time)
#define PHASES      4                         // async b128 ops in flight / wave

// bytes covered by one phase across the whole grid: 1024*256*16 = 4 MB
static constexpr uint64_t PHASE_STRIDE = (uint64_t)COPY_BLOCKS * COPY_BLOCK * 16u;
// bytes covered by one full sweep: 16 MB (every segment is a multiple)
static constexpr uint64_t SWEEP_BYTES  = PHASE_STRIDE * PHASES;

typedef unsigned u32x4 __attribute__((ext_vector_type(4)));
typedef unsigned u32x8 __attribute__((ext_vector_type(8)));

// ---------------------------------------------------------------------------
// Bulk streaming copy on the CDNA5 async DMA path (global -> LDS -> global).
// Static sweep geometry: no per-phase bounds checks at all. Inner loop is
// 4 unconditional async b128 loads (GVS: SGPR base + constant per-lane VGPR
// offset), s_wait_asynccnt 0, 4 async stores, s_wait_asynccnt 0, two scalar
// 64-bit adds and a 32-bit counted-loop branch. EXEC is all-ones throughout.
// ---------------------------------------------------------------------------
__global__ __launch_bounds__(COPY_BLOCK) void bulk_copy_async(
    const float* __restrict__ src, float* __restrict__ dst, unsigned iters) {
  __shared__ float4 lbuf[COPY_BLOCK * PHASES];

  const unsigned tid   = threadIdx.x;
  // Generic LDS pointers: [31:0] is the LDS byte offset (aperture in [63:32]).
  const unsigned lbase = (unsigned)(uintptr_t)(&lbuf[tid]);
  const unsigned voff  = tid * 16u;           // constant per-lane byte offset

  uint64_t sb = (uint64_t)(uintptr_t)src + (uint64_t)blockIdx.x * (COPY_BLOCK * 16u);
  uint64_t db = (uint64_t)(uintptr_t)dst + (uint64_t)blockIdx.x * (COPY_BLOCK * 16u);

  for (unsigned it = 0; it < iters; ++it) {   // uniform 32-bit counted loop
#pragma unroll
    for (int k = 0; k < PHASES; ++k) {
      asm volatile("global_load_async_to_lds_b128 %0, %1, %2"
                   :
                   : "v"(lbase + (unsigned)(k * COPY_BLOCK * 16)),
                     "v"(voff), "s"(sb + (uint64_t)k * PHASE_STRIDE)
                   : "memory");
    }
    asm volatile("s_wait_asynccnt 0" ::: "memory");   // LDS data resident
#pragma unroll
    for (int k = 0; k < PHASES; ++k) {
      asm volatile("global_store_async_from_lds_b128 %0, %1, %2"
                   :
                   : "v"(voff),
                     "v"(lbase + (unsigned)(k * COPY_BLOCK * 16)),
                     "s"(db + (uint64_t)k * PHASE_STRIDE)
                   : "memory");
    }
    asm volatile("s_wait_asynccnt 0" ::: "memory");   // LDS reusable
    sb += SWEEP_BYTES;
    db += SWEEP_BYTES;
  }
}

// ---------------------------------------------------------------------------
// TDM 2D tile store descriptor (groups 0 and 1 only; VADDR2/3 omitted = 2D).
// Packing per CDNA5 ISA 8.3/8.4:
//   g0: count=1 | lds_addr | global_addr[31:0] | global_addr[56:32],type=2
//   g1: data_size=2 (4B); tensor_dim0/1; tile_dim0/1 (tile_dim2=0 -> 2D);
//       tensor_dim0_stride = W elements.
// All inputs are wave-uniform, so the "s" constraints keep the D# in SGPRs.
// ---------------------------------------------------------------------------
__device__ __forceinline__ void tdm_store_2d(unsigned lds_addr, uint64_t gaddr,
                                             unsigned tensor_w, unsigned tensor_h,
                                             unsigned tile_w, unsigned tile_h,
                                             unsigned row_stride_elems) {
  u32x4 g0;
  g0.s0 = 1u;                                              // count=1, user mode
  g0.s1 = lds_addr;                                        // lds_addr[31:0]
  g0.s2 = (unsigned)gaddr;                                 // global_addr[31:0]
  g0.s3 = ((unsigned)(gaddr >> 32) & 0x01FFFFFFu) | (2u << 30); // addr[56:32]|type=2

  u32x8 g1;
  g1.s0 = 2u << 16;                                        // data_size=4B
  g1.s1 = (tensor_w & 0xFFFFu) << 16;                      // dim0[15:0] @63:48
  g1.s2 = (tensor_w >> 16) | ((tensor_h & 0xFFFFu) << 16); // dim0[31:16]|dim1[15:0]
  g1.s3 = (tensor_h >> 16) | (tile_w << 16);               // dim1[31:16]|tile_dim0
  g1.s4 = tile_h;                                          // tile_dim1; tile_dim2=0
  g1.s5 = row_stride_elems;                                // dim0_stride[31:0]
  g1.s6 = 0u;                                              // dim0_stride[47:32], dim1_stride[15:0]
  g1.s7 = 0u;                                              // dim1_stride[47:16]

  asm volatile("tensor_store_from_lds %0, %1" :: "s"(g0), "s"(g1) : "memory");
}

// ---------------------------------------------------------------------------
// Scatter one feature tile per wave via the Tensor Data Mover:
//   1) async-load the contiguous fdim*fdim*4B source tile into LDS (ASYNCcnt)
//   2) tensor_store_from_lds with a 2D D# targeting the dynamic
//      (hoff, woff) tile inside the H x W plane                    (TENSORcnt)
// One block = one wave = one (batch, channel) tile.
// ---------------------------------------------------------------------------
__global__ __launch_bounds__(32) void scatter_tdm(
    const float* __restrict__ feat, const int* __restrict__ Loc,
    float* __restrict__ dst, int cshift, int H, int W, int fdim,
    int locshift) {
  __shared__ float tile[64 * 64];                          // 16 KB max tile

  const int bc = (int)blockIdx.x;                          // b*C + c
  const int b  = bc >> cshift;
  const unsigned nbytes  = (unsigned)(fdim * fdim * 4);
  const unsigned lane    = threadIdx.x;
  const unsigned ldsbase = (unsigned)(uintptr_t)(&tile[0]);
  const uint64_t src     = (uint64_t)(uintptr_t)(feat + (size_t)bc * fdim * fdim);

  // Fill LDS with the source tile (wave32: 512B per sweep, b128 per lane).
  for (unsigned off = lane * 16u; off < nbytes; off += 512u) {
    asm volatile("global_load_async_to_lds_b128 %0, %1, %2"
                 :
                 : "v"(ldsbase + off), "v"(off), "s"(src)
                 : "memory");
  }
  asm volatile("s_wait_asynccnt 0" ::: "memory");

  const int woff = Loc[2 * b] >> locshift;
  const int hoff = Loc[2 * b + 1] >> locshift;
  const uint64_t gaddr = (uint64_t)(uintptr_t)(
      dst + ((size_t)bc * H + (size_t)hoff) * (size_t)W + (size_t)woff);

  tdm_store_2d(ldsbase, gaddr, (unsigned)W, (unsigned)H,
               (unsigned)fdim, (unsigned)fdim, (unsigned)W);
  __builtin_amdgcn_s_wait_tensorcnt(0);
}

// ---------------------------------------------------------------------------
extern "C" void kernel_launch(void* const* d_in, const int* in_sizes, int n_in,
                              void* d_out, int out_size, void* d_ws,
                              size_t ws_size, hipStream_t stream) {
  (void)in_sizes; (void)n_in; (void)out_size; (void)d_ws; (void)ws_size;

  const int*   Loc = (const int*)d_in[0];
  const float* f_b = (const float*)d_in[1];
  const float* f_3 = (const float*)d_in[2];
  const float* f_2 = (const float*)d_in[3];
  const float* f_1 = (const float*)d_in[4];
  const float* m_b = (const float*)d_in[5];
  const float* m_3 = (const float*)d_in[6];
  const float* m_2 = (const float*)d_in[7];
  const float* m_1 = (const float*)d_in[8];

  float* o_b = (float*)d_out;
  float* o_3 = o_b + N_B;
  float* o_2 = o_3 + N_3;
  float* o_1 = o_2 + N_2;

  auto copy = [&](const float* s, float* d, size_t nfloats) {
    unsigned iters = (unsigned)((nfloats * 4ull) / SWEEP_BYTES); // exact multiple
    bulk_copy_async<<<dim3(COPY_BLOCKS), dim3(COPY_BLOCK), 0, stream>>>(
        s, d, iters);
  };

  copy(m_b, o_b, N_B);   // 1 sweep
  copy(m_3, o_3, N_3);   // 2 sweeps
  copy(m_2, o_2, N_2);   // 4 sweeps
  copy(m_1, o_1, N_1);   // 16 sweeps

  auto scat = [&](const float* f, float* d, int C, int cshift, int H, int W,
                  int fdim, int locshift) {
    scatter_tdm<<<dim3(16 * C), dim3(32), 0, stream>>>(
        f, Loc, d, cshift, H, W, fdim, locshift);
  };

  // segment:      feat  out  C  csh   H    W  fdim locshift
  scat(f_b, o_b, 256, 8,  32,  32,  8, 4);
  scat(f_3, o_3, 128, 7,  64,  64, 16, 3);
  scat(f_2, o_2,  64, 6, 128, 128, 32, 2);
  scat(f_1, o_1,  64, 6, 256, 256, 64, 1);
}